// CliffordAlgebra_65128884077237
// MI455X (gfx1250) — compile-verified
//
#include <hip/hip_runtime.h>
#include <utility>

typedef __attribute__((ext_vector_type(2))) float v2f;
typedef __attribute__((ext_vector_type(4))) float v4f;
typedef __attribute__((ext_vector_type(8))) float v8f;

// Compile-time Cayley sign for Cl(3,1,0): metric e0..e2 = +1, e3 = -1.
constexpr int sign_ij(int i, int j) {
  int a = i >> 1, s = 0;
  while (a) { s += __builtin_popcount(a & j); a >>= 1; }
  int sg = (s & 1) ? -1 : 1;
  if (i & j & 8) sg = -sg;  // e3*e3 = -1
  return sg;
}

// Opaque per-lane-half select: lanes 0-15 get lo_val, lanes 16-31 get hi_val.
// Inline asm so clang cannot convert value-selects into dynamic array indexing.
__device__ __forceinline__ float half_select(float lo_val, float hi_val) {
  float r;
  unsigned m = 0xFFFFu;  // SGPR mask: bit set for lanes 0-15
  // v_cndmask_b32 dst, src0, src1, ssrc2 : dst = ssrc2[lane] ? src1 : src0
  asm("v_cndmask_b32 %0, %1, %2, %3" : "=v"(r) : "v"(hi_val), "v"(lo_val), "s"(m));
  return r;
}

// One V_WMMA_F32_16X16X4_F32 covering k = 16*II + 4*QQ + {0..3}
// (i = II, target columns c = 4*QQ + e). A-operand layout (16x4 f32):
//   lanes 0-15:  VGPR0 = K=0 (e=0), VGPR1 = K=1 (e=1)
//   lanes 16-31: VGPR0 = K=2 (e=2), VGPR1 = K=3 (e=3)
// Needed value: sign(i, u^2h)*a[i]*b[u^2h]  (u = i^4q, h = lane half)
//   bs[u]  == (lo ? b[u] : b[u^2])   (opaque cndmask pre-select)
//   Apm[i] == (lo ? a[i] : -a[i])    (single v_xor with per-lane sign mask)
template <int QQ, int II>
__device__ __forceinline__ void wmma_chunk(const float (&A)[16], const float (&Apm)[16],
                                           const float (&bs)[16], const v2f Bop, v8f& acc) {
  constexpr int u  = II ^ (4 * QQ);
  constexpr int s0 = sign_ij(II, u);       // lo lanes, A.x (e=0)
  constexpr int s2 = sign_ij(II, u ^ 2);   // hi lanes, A.x (e=2)
  constexpr int s1 = sign_ij(II, u ^ 1);   // lo lanes, A.y (e=1)
  constexpr int s3 = sign_ij(II, u ^ 3);   // hi lanes, A.y (e=3)
  float vx, vy;
  if constexpr (s0 == s2) vx = (s0 > 0 ? A[II]   : -A[II])   * bs[u];
  else                    vx = (s0 > 0 ? Apm[II] : -Apm[II]) * bs[u];
  if constexpr (s1 == s3) vy = (s1 > 0 ? A[II]   : -A[II])   * bs[u ^ 1];
  else                    vy = (s1 > 0 ? Apm[II] : -Apm[II]) * bs[u ^ 1];
  v2f Aop; Aop.x = vx; Aop.y = vy;
  acc = __builtin_amdgcn_wmma_f32_16x16x4_f32(
      /*neg_a=*/false, Aop, /*neg_b=*/false, Bop,
      /*c_mod=*/(short)0, acc, /*reuse_a=*/false, /*reuse_b=*/false);
}

// One q-group: constant one-hot B operand (4x16, row e one-hot at col 4q+e).
// B layout (4x16 f32): VGPR0 = rows K=0 (lanes 0-15) / K=2 (lanes 16-31),
//                      VGPR1 = rows K=1 / K=3.
template <int QQ, int... Is>
__device__ __forceinline__ void q_block(const float (&A)[16], const float (&Apm)[16],
                                        const float (&bs)[16], int tgt, v8f& acc,
                                        std::integer_sequence<int, Is...>) {
  v2f Bop;
  Bop.x = (tgt == 4 * QQ)     ? 1.0f : 0.0f;
  Bop.y = (tgt == 4 * QQ + 1) ? 1.0f : 0.0f;
  ((wmma_chunk<QQ, Is>(A, Apm, bs, Bop, acc)), ...);
}

__global__ __launch_bounds__(256) void clifford_gp_wmma(
    const float* __restrict__ a, const float* __restrict__ b,
    float* __restrict__ out, int nrows) {
  const int lane = threadIdx.x & 31;
  const int l    = lane & 15;   // column index / row-in-half
  const int h    = lane >> 4;   // lane half
  const int wave = blockIdx.x * (blockDim.x >> 5) + (threadIdx.x >> 5);

  const long r0 = (long)wave * 16;   // wave-uniform tile base
  if (r0 >= nrows) return;

  long myrow = r0 + l;                                // both halves load same 16 rows
  long ld = (myrow < nrows) ? myrow : (long)(nrows - 1);

  const v4f* arow = reinterpret_cast<const v4f*>(a + ld * 16);
  const v4f* brow = reinterpret_cast<const v4f*>(b + ld * 16);
  float A[16], Br[16];
#pragma unroll
  for (int t = 0; t < 4; ++t) {
    v4f av = arow[t], bv = brow[t];   // global_load_b128 x8, cachelines fully consumed
#pragma unroll
    for (int e = 0; e < 4; ++e) { A[4 * t + e] = av[e]; Br[4 * t + e] = bv[e]; }
  }

  // bs[u] = lo ? b[u] : b[u^2]  — one opaque v_cndmask each.
  float bs[16];
#pragma unroll
  for (int u = 0; u < 16; ++u) bs[u] = half_select(Br[u], Br[u ^ 2]);

  // Apm[i] = lo ? a[i] : -a[i] — single v_xor with per-lane sign-bit mask
  // (0 on lanes 0-15, 0x80000000 on lanes 16-31); integer form defeats any
  // select-to-index rewriting and needs no fneg materialization.
  const unsigned smask = (unsigned)(lane & 16) << 27;
  float Apm[16];
#pragma unroll
  for (int i = 0; i < 16; ++i)
    Apm[i] = __uint_as_float(__float_as_uint(A[i]) ^ smask);

  // One-hot target column within this lane's B rows: row e holds col 4q+e,
  // lo lanes carry rows e=0,1 / hi lanes rows e=2,3 -> match against l-2h.
  const int tgt = l - 2 * h;

  v8f acc = {0.f, 0.f, 0.f, 0.f, 0.f, 0.f, 0.f, 0.f};
  auto seq = std::make_integer_sequence<int, 16>{};
  q_block<0>(A, Apm, bs, tgt, acc, seq);
  q_block<1>(A, Apm, bs, tgt, acc, seq);
  q_block<2>(A, Apm, bs, tgt, acc, seq);
  q_block<3>(A, Apm, bs, tgt, acc, seq);

  // C/D layout: VGPR r holds rows r (lanes 0-15) and r+8 (lanes 16-31), col = l.
  // Epilogue: nan->0, +/-inf and overflow clipped to +/-1000 (matches reference).
#pragma unroll
  for (int r = 0; r < 8; ++r) {
    float v = acc[r];
    v = (v != v) ? 0.0f : v;
    v = fminf(fmaxf(v, -1000.0f), 1000.0f);
    long row = r0 + r + 8 * h;
    if (row < nrows) out[row * 16 + l] = v;
  }
}

extern "C" void kernel_launch(void* const* d_in, const int* in_sizes, int n_in,
                              void* d_out, int out_size, void* d_ws, size_t ws_size,
                              hipStream_t stream) {
  const float* a = (const float*)d_in[0];
  const float* b = (const float*)d_in[1];
  float* out = (float*)d_out;

  const int nrows = in_sizes[0] / 16;              // N multivectors
  const int ntiles = (nrows + 15) / 16;            // 16 rows per wave
  const int waves_per_block = 8;                   // 256 threads, wave32
  const int blocks = (ntiles + waves_per_block - 1) / waves_per_block;

  clifford_gp_wmma<<<blocks, 256, 0, stream>>>(a, b, out, nrows);
}